// GNNVotingModel_31653908971548
// MI455X (gfx1250) — compile-verified
//
#include <hip/hip_runtime.h>
#include <hip/hip_bf16.h>

#define FDIM 128
#define LDS_STRIDE 130   // 128 + 2 pad -> bank = (2*col + k) % 64, conflict-free

typedef __attribute__((ext_vector_type(2))) float v2f;
typedef __attribute__((ext_vector_type(8))) float v8f;

// ---------------------------------------------------------------- small utils
__global__ void k_fill1(float* __restrict__ p, int n) {
    int i = blockIdx.x * blockDim.x + threadIdx.x;
    if (i < n) p[i] = 1.0f;                       // self-loop contributes 1 to deg
}

__global__ void k_deg(const int* __restrict__ dst, float* deg, int e) {
    int i = blockIdx.x * blockDim.x + threadIdx.x;
    if (i < e) atomicAdd(&deg[dst[i]], 1.0f);
}

__global__ void k_rsqrt(const float* __restrict__ deg, float* __restrict__ dinv, int n) {
    int i = blockIdx.x * blockDim.x + threadIdx.x;
    if (i < n) dinv[i] = rsqrtf(deg[i]);          // deg >= 1 always (self loops)
}

// ------------------------------------------------- fused GEMM + self-loop init
// H   = A @ W                (n x 128) = (n x 128)(128 x 128), fp32 WMMA
// AGG = dinv^2 * H           (self-loop contribution, initializes accumulator)
// One wave computes a 16x128 row tile; 8 waves/block -> 128 rows/block.
// A-fragment addresses for out-of-range rows are CLAMPED to row n-1 (always
// valid memory) instead of predicated: those rows only feed accumulator rows
// that are masked out at the store, so no exec-mask dance in the hot loop.
// NOTE: A and AGG may alias (layer 2 runs in place). Each wave only reads its
// own 16 rows of A and stores to the same 16 rows of AGG strictly afterwards,
// so this is safe; A/H/AGG deliberately NOT marked __restrict__.
__global__ __launch_bounds__(256) void k_gcn_gemm(
    const float* A, const float* __restrict__ W,
    const float* __restrict__ dinv, float* H, float* AGG, int n)
{
    __shared__ float Wl[FDIM * LDS_STRIDE];

    // cooperative load of W into LDS, transposed: Wl[col*130 + k] = W[k][col]
    for (int idx = threadIdx.x; idx < FDIM * FDIM; idx += 256) {
        int k = idx >> 7;
        int c = idx & (FDIM - 1);
        Wl[c * LDS_STRIDE + k] = W[idx];
    }
    __syncthreads();

    const int wave = threadIdx.x >> 5;
    const int lane = threadIdx.x & 31;
    const int half = lane >> 4;          // 0: lanes 0-15, 1: lanes 16-31
    const int lm   = lane & 15;
    const int rowBase = blockIdx.x * 128 + wave * 16;

    // A fragment source: lane half h reads K = kk+2h, kk+2h+1 of row rowBase+lm
    const int arow = min(rowBase + lm, n - 1);   // clamp: no predication needed
    const float* aPtr = A + (size_t)arow * FDIM + 2 * half;

    v8f acc[8] = {};   // 8 N-tiles x 8 VGPRs = full 16x128 tile

    #pragma unroll 4
    for (int kk = 0; kk < FDIM; kk += 4) {
        v2f a = *(const v2f*)(aPtr + kk);        // 8B aligned global_load_b64
        #pragma unroll
        for (int t = 0; t < 8; ++t) {
            const int col = t * 16 + lm;
            v2f b = *(const v2f*)&Wl[col * LDS_STRIDE + kk + 2 * half]; // ds_load_b64
            acc[t] = __builtin_amdgcn_wmma_f32_16x16x4_f32(
                false, a, false, b, (short)0, acc[t], false, false);
        }
    }

    // C/D layout: VGPR r -> M = r + 8*half, lane lm -> N column within tile
    #pragma unroll
    for (int r = 0; r < 8; ++r) {
        const int row = rowBase + r + 8 * half;
        if (row < n) {
            const float dv = dinv[row];
            const float w2 = dv * dv;
            float* hrow = H   + (size_t)row * FDIM;
            float* grow = AGG + (size_t)row * FDIM;
            #pragma unroll
            for (int t = 0; t < 8; ++t) {
                const float v = acc[t][r];
                const int   c = t * 16 + lm;
                hrow[c] = v;
                grow[c] = w2 * v;      // fused self-loop init of accumulator
            }
        }
    }
}

// ------------------------------------------------------- edge scatter (128 f)
// One wave per edge: coalesced 512B gather of h[src], 4 f32 atomics/lane.
// Edge id written as an explicitly wave-uniform expression so the scalar
// unit handles src/dst/dinv loads.
__global__ __launch_bounds__(256) void k_edge_scatter(
    const float* __restrict__ H,
    const int* __restrict__ src,
    const int* __restrict__ dst,
    const float* __restrict__ dinv,
    float* AGG, int e)
{
    int ed = blockIdx.x * 8 + (threadIdx.x >> 5);   // 8 waves per block
    if (ed >= e) return;
    int lane = threadIdx.x & 31;
    int s = src[ed], d = dst[ed];
    float w = dinv[s] * dinv[d];
    const float4* hp = (const float4*)(H + (size_t)s * FDIM);
    float4 v = hp[lane];
    float* ap = AGG + (size_t)d * FDIM + lane * 4;
    atomicAdd(ap + 0, w * v.x);
    atomicAdd(ap + 1, w * v.y);
    atomicAdd(ap + 2, w * v.z);
    atomicAdd(ap + 3, w * v.w);
}

__global__ void k_bias_relu(float* A, const float* __restrict__ b, int total) {
    int i = blockIdx.x * blockDim.x + threadIdx.x;
    if (i < total) {
        float v = A[i] + b[i & (FDIM - 1)];
        A[i] = fmaxf(v, 0.0f);
    }
}

// --------------------------------------------- layer 3: h3 = X @ W3 (128 -> 1)
// wave per node; fused self-loop init of agg3.
__global__ void k_dot_w3(const float* __restrict__ X, const float* __restrict__ W3,
                         const float* __restrict__ dinv,
                         float* __restrict__ h3, float* __restrict__ agg3, int n)
{
    int node = blockIdx.x * 8 + (threadIdx.x >> 5);
    if (node >= n) return;
    int lane = threadIdx.x & 31;
    const float4* xp = (const float4*)(X + (size_t)node * FDIM);
    const float4* wp = (const float4*)W3;
    float4 xv = xp[lane];
    float4 wv = wp[lane];
    float s = xv.x * wv.x + xv.y * wv.y + xv.z * wv.z + xv.w * wv.w;
    #pragma unroll
    for (int m = 16; m > 0; m >>= 1) s += __shfl_xor(s, m, 32);
    if (lane == 0) {
        h3[node] = s;
        float dv = dinv[node];
        agg3[node] = dv * dv * s;
    }
}

__global__ void k_edge_scalar(const float* __restrict__ h3,
                              const int* __restrict__ src,
                              const int* __restrict__ dst,
                              const float* __restrict__ dinv,
                              float* agg3, int e)
{
    int i = blockIdx.x * blockDim.x + threadIdx.x;
    if (i < e) {
        int s = src[i], d = dst[i];
        atomicAdd(&agg3[d], dinv[s] * dinv[d] * h3[s]);
    }
}

__global__ void k_sigmoid(const float* __restrict__ agg3, const float* __restrict__ b3,
                          float* __restrict__ out, int n)
{
    int i = blockIdx.x * blockDim.x + threadIdx.x;
    if (i < n) {
        float v = agg3[i] + b3[0];
        out[i] = 1.0f / (1.0f + __expf(-v));
    }
}

// ---------------------------------------------------------------------- launch
static inline int cdiv(long long a, long long b) { return (int)((a + b - 1) / b); }

extern "C" void kernel_launch(void* const* d_in, const int* in_sizes, int n_in,
                              void* d_out, int out_size, void* d_ws, size_t ws_size,
                              hipStream_t stream) {
    const float* x  = (const float*)d_in[0];
    const int*   ei = (const int*)  d_in[1];
    const float* W1 = (const float*)d_in[2];
    const float* b1 = (const float*)d_in[3];
    const float* W2 = (const float*)d_in[4];
    const float* b2 = (const float*)d_in[5];
    const float* W3 = (const float*)d_in[6];
    const float* b3 = (const float*)d_in[7];
    float* out = (float*)d_out;

    const int n = in_sizes[0] / FDIM;
    const int e = in_sizes[1] / 2;
    const int* src = ei;
    const int* dst = ei + e;

    float* ws   = (float*)d_ws;
    float* deg  = ws;
    float* dinv = ws + (size_t)n;
    float* h3   = ws + 2 * (size_t)n;
    float* agg3 = ws + 3 * (size_t)n;
    float* bufH = ws + 4 * (size_t)n;                 // n x 128
    float* bufA = bufH + (size_t)n * FDIM;            // n x 128

    const int T = 256;

    // degrees + symmetric norm
    k_fill1 <<<cdiv(n, T), T, 0, stream>>>(deg, n);
    k_deg   <<<cdiv(e, T), T, 0, stream>>>(dst, deg, e);
    k_rsqrt <<<cdiv(n, T), T, 0, stream>>>(deg, dinv, n);

    // layer 1
    k_gcn_gemm    <<<cdiv(n, 128), T, 0, stream>>>(x, W1, dinv, bufH, bufA, n);
    k_edge_scatter<<<cdiv(e, 8), T, 0, stream>>>(bufH, src, dst, dinv, bufA, e);
    k_bias_relu   <<<cdiv((long long)n * FDIM, T), T, 0, stream>>>(bufA, b1, n * FDIM);

    // layer 2 (in place: A == AGG == bufA, safe per-wave row ownership)
    k_gcn_gemm    <<<cdiv(n, 128), T, 0, stream>>>(bufA, W2, dinv, bufH, bufA, n);
    k_edge_scatter<<<cdiv(e, 8), T, 0, stream>>>(bufH, src, dst, dinv, bufA, e);
    k_bias_relu   <<<cdiv((long long)n * FDIM, T), T, 0, stream>>>(bufA, b2, n * FDIM);

    // layer 3 (128 -> 1) + sigmoid
    k_dot_w3      <<<cdiv(n, 8), T, 0, stream>>>(bufA, W3, dinv, h3, agg3, n);
    k_edge_scalar <<<cdiv(e, T), T, 0, stream>>>(h3, src, dst, dinv, agg3, e);
    k_sigmoid     <<<cdiv(n, T), T, 0, stream>>>(agg3, b3, out, n);
}